// transforms_blue_78340203479774
// MI455X (gfx1250) — compile-verified
//
#include <hip/hip_runtime.h>
#include <stdint.h>

// transforms_blue for MI455X (gfx1250).
//
// HBM-streaming elementwise kernel: 402.7 MB traffic -> ~17.3 us floor at
// 23.3 TB/s. No matrix structure -> no WMMA. CDNA5 paths used instead:
//   * global_load_async_to_lds_b128 (ASYNCcnt) double-buffered pipeline:
//     loads don't hold VGPRs in flight, 1-tile lead distance per wave.
//   * s_wait_asynccnt split-counter waits (in-order async-load completion).
//   * Non-temporal b128 stores (each 201 MB stream exceeds the 192 MB L2).
//   * wave32-friendly 256-thread blocks (8 waves), ds_load_b128 readback.

typedef float v4f __attribute__((ext_vector_type(4)));

#define S4        65536              // 512*512/4 float4 per channel plane
#define PLANES3   (3 * S4)           // float4 per image (3 planes)
#define NTHREADS  256
#define NBLOCKS   4096
#define NTILES    4                  // 4096*256*4 == 4,194,304 float4-pixels
#define TILE_B    (NTHREADS * 16)    // bytes per plane per buffer (4096)
#define BUF_B     (3 * TILE_B)       // bytes per buffer (12288)

__device__ __forceinline__ void blue_math(const v4f u0, const v4f u1, const v4f u2,
                                          v4f& c0, v4f& c1, v4f& c2)
{
#pragma unroll
    for (int i = 0; i < 4; ++i) {
        // uint8 cast of non-negative input == floor(x*255)
        const float f0 = floorf(u0[i] * 255.0f);
        const float f1 = floorf(u1[i] * 255.0f);
        const float f2 = floorf(u2[i] * 255.0f);
        // cv2 BGR2GRAY weights; jnp.round == round-half-even == rintf (RNE mode)
        const float gray = rintf(0.114f * f0 + 0.587f * f1 + 0.299f * f2);

        const bool  m  = gray < 128.0f;
        const float a0 = fminf(gray + 60.0f, 255.0f);
        const float a1 = fminf(gray + 10.0f, 255.0f);

        const float inv255 = 1.0f / 255.0f;
        c0[i] = (m ? a0 : gray) * inv255;
        c1[i] = (m ? a1 : gray) * inv255;
        c2[i] = gray * inv255;
    }
}

// ---------------------------------------------------------------------------
// Main path: async global->LDS double-buffered pipeline. Requires the exact
// harness shape so every wave runs EXACTLY NTILES iterations with EXEC all-1s
// (wave-uniform ASYNCcnt bookkeeping).
// ---------------------------------------------------------------------------
__global__ __launch_bounds__(NTHREADS, 1)
void transforms_blue_async(const float* __restrict__ in,
                           float* __restrict__ out,
                           int stride4)   // = NBLOCKS*NTHREADS
{
    __shared__ __align__(16) unsigned char lds[2 * BUF_B];

    const int tid = threadIdx.x;
    // Generic pointer to __shared__ carries the LDS allocation offset in its
    // low 32 bits (ISA 10.2) -> exactly the async instruction's LDS address.
    const uint32_t lds0 = (uint32_t)(uintptr_t)(&lds[0]) + (uint32_t)tid * 16u;

    const v4f* __restrict__ pin  = (const v4f*)in;
    v4f*       __restrict__ pout = (v4f*)out;

    const int idx0 = blockIdx.x * NTHREADS + tid;

    // Issue one tile's three plane loads (ASYNCcnt += 3) into buffer `buf`.
    auto issue = [&](int idx, uint32_t buf) {
        const int  b    = idx >> 16;          // image
        const int  s4   = idx & (S4 - 1);     // spatial float4
        const v4f* g    = pin + (long)b * PLANES3 + s4;
        const uint32_t l = lds0 + buf * (uint32_t)BUF_B;
        asm volatile("global_load_async_to_lds_b128 %0, %1, off th:TH_LOAD_NT"
                     :: "v"(l),              "v"(g)          : "memory");
        asm volatile("global_load_async_to_lds_b128 %0, %1, off th:TH_LOAD_NT"
                     :: "v"(l + TILE_B),     "v"(g + S4)     : "memory");
        asm volatile("global_load_async_to_lds_b128 %0, %1, off th:TH_LOAD_NT"
                     :: "v"(l + 2 * TILE_B), "v"(g + 2 * S4) : "memory");
    };

    issue(idx0, 0u);   // prologue: tile 0 -> buffer 0

#pragma unroll
    for (int t = 0; t < NTILES; ++t) {
        const int cur = idx0 + t * stride4;

        if (t < NTILES - 1) {
            issue(cur + stride4, (uint32_t)((t + 1) & 1));   // next tile -> other buffer
            // Async loads complete in order: <=3 outstanding means tile t is in LDS.
            asm volatile("s_wait_asynccnt 0x3" ::: "memory");
        } else {
            asm volatile("s_wait_asynccnt 0x0" ::: "memory");
        }

        const uint32_t buf = (uint32_t)(t & 1);
        const unsigned char* lb = &lds[buf * BUF_B + tid * 16];
        const v4f u0 = *(const v4f*)(lb);                 // ds_load_b128
        const v4f u1 = *(const v4f*)(lb + TILE_B);
        const v4f u2 = *(const v4f*)(lb + 2 * TILE_B);

        v4f c0, c1, c2;
        blue_math(u0, u1, u2, c0, c1, c2);

        const long base = (long)(cur >> 16) * PLANES3 + (cur & (S4 - 1));
        __builtin_nontemporal_store(c0, pout + base);
        __builtin_nontemporal_store(c1, pout + base + S4);
        __builtin_nontemporal_store(c2, pout + base + 2 * S4);
    }
}

// ---------------------------------------------------------------------------
// Generic fallback (any size): direct NT b128 loads, same math.
// ---------------------------------------------------------------------------
__global__ __launch_bounds__(NTHREADS, 1)
void transforms_blue_direct(const float* __restrict__ in,
                            float* __restrict__ out,
                            int total4)
{
    const int idx = blockIdx.x * NTHREADS + threadIdx.x;
    if (idx >= total4) return;

    const v4f* __restrict__ pin  = (const v4f*)in;
    v4f*       __restrict__ pout = (v4f*)out;

    const long base = (long)(idx >> 16) * PLANES3 + (idx & (S4 - 1));

    const v4f u0 = __builtin_nontemporal_load(pin + base);
    const v4f u1 = __builtin_nontemporal_load(pin + base + S4);
    const v4f u2 = __builtin_nontemporal_load(pin + base + 2 * S4);

    v4f c0, c1, c2;
    blue_math(u0, u1, u2, c0, c1, c2);

    __builtin_nontemporal_store(c0, pout + base);
    __builtin_nontemporal_store(c1, pout + base + S4);
    __builtin_nontemporal_store(c2, pout + base + 2 * S4);
}

extern "C" void kernel_launch(void* const* d_in, const int* in_sizes, int n_in,
                              void* d_out, int out_size, void* d_ws, size_t ws_size,
                              hipStream_t stream)
{
    (void)n_in; (void)out_size; (void)d_ws; (void)ws_size;

    const float* in  = (const float*)d_in[0];
    float*       out = (float*)d_out;

    // in_sizes[0] = 64*3*512*512 floats; one work item = one float4 across 3 planes
    const int total4  = in_sizes[0] / 12;                 // 4,194,304 for the harness shape
    const int stride4 = NBLOCKS * NTHREADS;               // 1,048,576

    if (total4 == stride4 * NTILES) {
        transforms_blue_async<<<NBLOCKS, NTHREADS, 0, stream>>>(in, out, stride4);
    } else {
        const int blocks = (total4 + NTHREADS - 1) / NTHREADS;
        transforms_blue_direct<<<blocks, NTHREADS, 0, stream>>>(in, out, total4);
    }
}